// HyperbolicGCN_52742198395105
// MI455X (gfx1250) — compile-verified
//
#include <hip/hip_runtime.h>
#include <math.h>

// Hyperbolic GCN (LightGCN on the Poincare ball) for MI455X / gfx1250.
// Memory/atomic-bound sparse message passing.
// CDNA5 features used: GLOBAL_LOAD_ASYNC_TO_LDS_B64 (ASYNCcnt) for an
// 8-deep pipelined row gather into LDS, S_WAIT_ASYNCCNT staged waits,
// wave32 shuffle reductions, non-returning global_atomic_add_f32 scatter.

#define D 64            // embedding dim
#define WAVE 32         // gfx1250 is wave32
#define WPB 8           // waves per block
#define BLOCK (WAVE * WPB)
#define N_LAYERS 2
#define EPW 8           // edges per wave (async pipeline depth)

// ---- wave32 helpers -------------------------------------------------------

__device__ __forceinline__ float wave_sum(float v) {
#pragma unroll
  for (int m = 16; m > 0; m >>= 1) v += __shfl_xor(v, m, 32);
  return v;
}

__device__ __forceinline__ float softmax_w(const float* __restrict__ lw, int i) {
  float a = lw[0], b = lw[1], c = lw[2];
  float m = fmaxf(a, fmaxf(b, c));
  float ea = __expf(a - m), eb = __expf(b - m), ec = __expf(c - m);
  float s = ea + eb + ec;
  float w = (i == 0) ? ea : ((i == 1) ? eb : ec);
  return w / s;
}

// ---- kernel 1: h = project(exp_map(concat(ue, ie))); out = w0 * h ---------

__global__ void hgcn_init(const float* __restrict__ ue, const float* __restrict__ ie,
                          const float* __restrict__ lw,
                          float* __restrict__ h, float* __restrict__ out,
                          int num_users, int N) {
  int wid = blockIdx.x * WPB + (threadIdx.x >> 5);
  if (wid >= N) return;
  int lane = threadIdx.x & 31;

  const float* src = (wid < num_users) ? (ue + (size_t)wid * D)
                                       : (ie + (size_t)(wid - num_users) * D);
  float2 v = *(const float2*)(src + 2 * lane);

  // exp_map (c = 1)
  float n  = sqrtf(wave_sum(v.x * v.x + v.y * v.y));
  float nm = fmaxf(n, 1e-8f);
  float t  = tanhf(nm) / nm;
  float yx = v.x * t, yy = v.y * t;

  // project
  float ny = sqrtf(wave_sum(yx * yx + yy * yy));
  const float maxn = 1.0f - 1e-5f;
  if (ny > maxn) {
    float s = maxn / fmaxf(ny, 1e-12f);
    yx *= s; yy *= s;
  }

  size_t o = (size_t)wid * D + 2 * lane;
  *(float2*)(h + o) = make_float2(yx, yy);

  float w0 = softmax_w(lw, 0);
  *(float2*)(out + o) = make_float2(w0 * yx, w0 * yy);
}

// ---- kernel 2: h <- log_map(h) in place -----------------------------------

__global__ void hgcn_log(float* __restrict__ h, int N) {
  int wid = blockIdx.x * WPB + (threadIdx.x >> 5);
  if (wid >= N) return;
  int lane = threadIdx.x & 31;

  size_t o = (size_t)wid * D + 2 * lane;
  float2 v = *(const float2*)(h + o);

  float n = sqrtf(wave_sum(v.x * v.x + v.y * v.y));
  n = fminf(fmaxf(n, 1e-8f), 1.0f - 1e-5f);
  float t = (0.5f * logf((1.0f + n) / (1.0f - n))) / n;  // atanh(n)/n

  *(float2*)(h + o) = make_float2(v.x * t, v.y * t);
}

// ---- kernel 3: zero the aggregation buffer --------------------------------

__global__ void hgcn_zero(float4* __restrict__ p, long n4) {
  long i = (long)blockIdx.x * blockDim.x + threadIdx.x;
  long stride = (long)gridDim.x * blockDim.x;
  for (; i < n4; i += stride) p[i] = make_float4(0.f, 0.f, 0.f, 0.f);
}

// ---- kernel 4: gather + scale + scatter-add, 8-deep async pipeline --------
// agg[row] += val * ht[col]   (one wave owns EPW edges; each 256B row is
// gathered via GLOBAL_LOAD_ASYNC_TO_LDS_B64, consumed with staged
// S_WAIT_ASYNCCNT so up to 8 gathers stay in flight per wave)

__global__ void __launch_bounds__(BLOCK)
hgcn_scatter(const float* __restrict__ vals,
             const int*   __restrict__ rows,
             const int*   __restrict__ cols,
             const float* __restrict__ ht,
             float*       __restrict__ agg,
             int nnz) {
  __shared__ float lds[WPB][EPW][D];    // 8 waves * 8 edges * 256B = 16KB

  int wave = threadIdx.x >> 5;
  int lane = threadIdx.x & 31;
  long eb = ((long)blockIdx.x * WPB + wave) * EPW;
  if (eb >= nnz) return;                 // wave-uniform

  // lanes 0..EPW-1 hold this wave's edge descriptors (clamped for tail)
  int myc = 0, myr = 0; float mya = 0.f;
  if (lane < EPW) {
    long el = eb + lane;
    if (el > nnz - 1) el = nnz - 1;      // clamp: duplicate gather, atomic masked later
    myc = cols[el];
    myr = rows[el];
    mya = vals[el];
  }

  // LDS byte address of this wave's staging area (low 32 bits of flat addr)
  unsigned lbase = (unsigned)(size_t)(&lds[wave][0][0]);

  // issue EPW async row-gathers: 32 lanes x 8B = one 256B row per op
#pragma unroll
  for (int i = 0; i < EPW; ++i) {
    int ci = __shfl(myc, i, 32);
    unsigned long long ga =
        (unsigned long long)(size_t)(ht + (size_t)ci * D) + 8ull * (unsigned)lane;
    unsigned la = lbase + (unsigned)(i * D * 4) + (unsigned)lane * 8u;
    asm volatile("global_load_async_to_lds_b64 %0, %1, off"
                 :: "v"(la), "v"(ga) : "memory");
  }

  // consume in order; async loads retire in order, so after waiting
  // ASYNCcnt <= (EPW-1-i) rows 0..i are resident in LDS.
#define HGCN_STEP(I, CNT)                                                   \
  {                                                                         \
    asm volatile("s_wait_asynccnt " #CNT ::: "memory");                     \
    float2 v = *(const float2*)&lds[wave][I][2 * lane];                     \
    int   ri = __shfl(myr, I, 32);                                          \
    float ai = __shfl(mya, I, 32);                                          \
    if ((eb + I) < nnz) {                                                   \
      float* dst = agg + (size_t)ri * D + 2 * lane;                         \
      atomicAdd(dst,     ai * v.x);  /* global_atomic_add_f32, no return */ \
      atomicAdd(dst + 1, ai * v.y);                                         \
    }                                                                       \
  }
  HGCN_STEP(0, 7)
  HGCN_STEP(1, 6)
  HGCN_STEP(2, 5)
  HGCN_STEP(3, 4)
  HGCN_STEP(4, 3)
  HGCN_STEP(5, 2)
  HGCN_STEP(6, 1)
  HGCN_STEP(7, 0)
#undef HGCN_STEP
}

// ---- kernel 5: h = project(exp_map(agg)); out += w[l] * h -----------------

__global__ void hgcn_expproj(const float* __restrict__ agg, const float* __restrict__ lw,
                             float* __restrict__ h, float* __restrict__ out,
                             int N, int widx) {
  int wid = blockIdx.x * WPB + (threadIdx.x >> 5);
  if (wid >= N) return;
  int lane = threadIdx.x & 31;

  size_t o = (size_t)wid * D + 2 * lane;
  float2 v = *(const float2*)(agg + o);

  float n  = sqrtf(wave_sum(v.x * v.x + v.y * v.y));
  float nm = fmaxf(n, 1e-8f);
  float t  = tanhf(nm) / nm;
  float yx = v.x * t, yy = v.y * t;

  float ny = sqrtf(wave_sum(yx * yx + yy * yy));
  const float maxn = 1.0f - 1e-5f;
  if (ny > maxn) {
    float s = maxn / fmaxf(ny, 1e-12f);
    yx *= s; yy *= s;
  }

  *(float2*)(h + o) = make_float2(yx, yy);

  float w = softmax_w(lw, widx);
  float2 acc = *(const float2*)(out + o);
  acc.x += w * yx;
  acc.y += w * yy;
  *(float2*)(out + o) = acc;
}

// ---- kernel 6: out <- project(out) in place -------------------------------

__global__ void hgcn_final(float* __restrict__ out, int N) {
  int wid = blockIdx.x * WPB + (threadIdx.x >> 5);
  if (wid >= N) return;
  int lane = threadIdx.x & 31;

  size_t o = (size_t)wid * D + 2 * lane;
  float2 v = *(const float2*)(out + o);

  float n = sqrtf(wave_sum(v.x * v.x + v.y * v.y));
  const float maxn = 1.0f - 1e-5f;
  if (n > maxn) {
    float s = maxn / fmaxf(n, 1e-12f);
    v.x *= s; v.y *= s;
    *(float2*)(out + o) = v;
  }
}

// ---- launch ---------------------------------------------------------------

extern "C" void kernel_launch(void* const* d_in, const int* in_sizes, int n_in,
                              void* d_out, int out_size, void* d_ws, size_t ws_size,
                              hipStream_t stream) {
  const float* ue   = (const float*)d_in[0];
  const float* ie   = (const float*)d_in[1];
  const float* lw   = (const float*)d_in[2];
  const float* vals = (const float*)d_in[3];
  const int*   rows = (const int*)d_in[4];
  const int*   cols = (const int*)d_in[5];
  // d_in[6] = num_users_split (scalar); equals in_sizes[0]/D, derived below.

  int num_users = in_sizes[0] / D;
  int num_items = in_sizes[1] / D;
  int N   = num_users + num_items;
  int nnz = in_sizes[3];

  float* h   = (float*)d_ws;                 // [N*D] tangent/ball embeddings
  float* agg = h + (size_t)N * D;            // [N*D] scatter-add target
  float* out = (float*)d_out;                // [N*D] weighted accumulator -> final

  int rowBlocks  = (N + WPB - 1) / WPB;
  long edgesPerBlock = (long)WPB * EPW;
  int edgeBlocks = (int)((nnz + edgesPerBlock - 1) / edgesPerBlock);
  long n4 = ((long)N * D) / 4;
  long zb = (n4 + BLOCK - 1) / BLOCK;
  int zeroBlocks = (int)(zb < 8192 ? zb : 8192);

  hgcn_init<<<rowBlocks, BLOCK, 0, stream>>>(ue, ie, lw, h, out, num_users, N);
  for (int l = 1; l <= N_LAYERS; ++l) {
    hgcn_log<<<rowBlocks, BLOCK, 0, stream>>>(h, N);
    hgcn_zero<<<zeroBlocks, BLOCK, 0, stream>>>((float4*)agg, n4);
    hgcn_scatter<<<edgeBlocks, BLOCK, 0, stream>>>(vals, rows, cols, h, agg, nnz);
    hgcn_expproj<<<rowBlocks, BLOCK, 0, stream>>>(agg, lw, h, out, N, l);
  }
  hgcn_final<<<rowBlocks, BLOCK, 0, stream>>>(out, N);
}